// Model_89704686944630
// MI455X (gfx1250) — compile-verified
//
#include <hip/hip_runtime.h>

// Jacobi 5-point stencil (edge-clamped), 10 steps fused via LDS temporal blocking.
// grid: (8,4,1024,1024) fp32 -> 32 images of 1024x1024.
// Tile: 64x64 output, halo 10 -> 84x84 input tile in LDS (+2 column shift so the
// output sub-tile and compute quads are 16B-aligned).
// CDNA5 data paths:
//   fill : global_load_async_to_lds_b64 (interior) / _b32 (border), ASYNCcnt
//   comp : 4-wide register blocking, ds_load_b128/ds_store_b128
//   drain: global_store_async_from_lds_b128

#define ALPHA   0.25f
#define NSTEPS  10
#define BX      64
#define HALO    NSTEPS
#define TILE    (BX + 2 * HALO)   // 84
#define CSH     2                 // column shift: quad cols (qx≡2 mod 4)+CSH are 16B aligned
#define TSTR    (TILE + 4)        // 88 floats row stride (352B, multiple of 16)
#define W       1024
#define H       1024

typedef float v4f __attribute__((ext_vector_type(4)));
typedef int v2i __attribute__((ext_vector_type(2)));
typedef int v4i __attribute__((ext_vector_type(4)));
typedef __attribute__((address_space(1))) int g_int;
typedef __attribute__((address_space(3))) int l_int;
typedef __attribute__((address_space(1))) v2i g_v2i;
typedef __attribute__((address_space(3))) v2i l_v2i;
typedef __attribute__((address_space(1))) v4i g_v4i;
typedef __attribute__((address_space(3))) v4i l_v4i;

#if __has_builtin(__builtin_amdgcn_global_load_async_to_lds_b32)
#define HAVE_AL32 1
#else
#define HAVE_AL32 0
#endif
#if __has_builtin(__builtin_amdgcn_global_load_async_to_lds_b64)
#define HAVE_AL64 1
#else
#define HAVE_AL64 0
#endif
#if __has_builtin(__builtin_amdgcn_global_store_async_from_lds_b128)
#define HAVE_AS128 1
#else
#define HAVE_AS128 0
#endif

__device__ __forceinline__ void wait_async_zero() {
#if __has_builtin(__builtin_amdgcn_s_wait_asynccnt)
  __builtin_amdgcn_s_wait_asynccnt(0);
#else
  asm volatile("s_wait_asynccnt 0" ::: "memory");
#endif
}

__global__ __launch_bounds__(256) void jacobi10_fused(const float* __restrict__ grid,
                                                      float* __restrict__ out) {
  __shared__ __align__(16) float bufA[TILE][TSTR];
  __shared__ __align__(16) float bufB[TILE][TSTR];

  const int tid = threadIdx.x;
  const int bgx = (int)blockIdx.x * BX - HALO;   // global x of local lx=0
  const int bgy = (int)blockIdx.y * BX - HALO;   // global y of local ly=0
  const float* img  = grid + (size_t)blockIdx.z * (size_t)(W * H);
  float*       oimg = out  + (size_t)blockIdx.z * (size_t)(W * H);

  const bool border = (bgx <= 0) | (bgy <= 0) | (bgx + TILE >= W) | (bgy + TILE >= H);

  // ---- Phase 1: fill 84x84 tile into bufA at column base CSH ----
  if (!border) {
#if HAVE_AL64
    for (int i = tid; i < TILE * 42; i += 256) {   // 42 b64 chunks per row
      int cx = (i % 42) * 2;
      int ly = i / 42;
      __builtin_amdgcn_global_load_async_to_lds_b64(
          (g_v2i*)(img + (size_t)(bgy + ly) * W + (bgx + cx)),
          (l_v2i*)&bufA[ly][cx + CSH], 0, 0);
    }
    wait_async_zero();
#else
    for (int i = tid; i < TILE * TILE; i += 256) {
      int lx = i % TILE, ly = i / TILE;
      bufA[ly][lx + CSH] = img[(size_t)(bgy + ly) * W + (bgx + lx)];
    }
#endif
  } else {
    for (int i = tid; i < TILE * TILE; i += 256) {
      int lx = i % TILE, ly = i / TILE;
      int gx = min(max(bgx + lx, 0), W - 1);
      int gy = min(max(bgy + ly, 0), H - 1);
#if HAVE_AL32
      __builtin_amdgcn_global_load_async_to_lds_b32(
          (g_int*)(img + (size_t)gy * W + gx),
          (l_int*)&bufA[ly][lx + CSH], 0, 0);
#else
      bufA[ly][lx + CSH] = img[(size_t)gy * W + gx];
#endif
    }
#if HAVE_AL32
    wait_async_zero();
#endif
  }
  __syncthreads();

  // ---- Phase 2: 10 Jacobi steps, ping-pong in LDS ----
  float (*src)[TSTR] = bufA;
  float (*dst)[TSTR] = bufB;

  if (!border) {
    // Vectorized interior path: each thread computes a 16B-aligned quad of 4 x.
    // Quads anchored at qx ≡ 2 (mod 4); lanes outside [lo,hi) produce garbage
    // that lands only in stale-region / padding LDS cells (never read again).
    const int qlane = (tid & 7) * 4;   // 8 quads across = 32 columns
    const int rlane = tid >> 3;        // 32 rows
    for (int s = 0; s < NSTEPS; ++s) {
      const int lo = s + 1, hi = TILE - (s + 1);
      const int qbase = ((lo - 2) & ~3) + 2;        // ≡ 2 mod 4, ≤ lo
      for (int ly = lo + rlane; ly < hi; ly += 32) {
        for (int qx = qbase + qlane; qx < hi; qx += 32) {
          const v4f c4 = *(const v4f*)&src[ly][qx + CSH];
          const v4f u4 = *(const v4f*)&src[ly - 1][qx + CSH];
          const v4f d4 = *(const v4f*)&src[ly + 1][qx + CSH];
          const float lm = src[ly][qx - 1 + CSH];
          const float rp = src[ly][qx + 4 + CSH];
          const v4f l4 = {lm, c4.x, c4.y, c4.z};
          const v4f r4 = {c4.y, c4.z, c4.w, rp};
          *(v4f*)&dst[ly][qx + CSH] =
              c4 + ALPHA * (l4 + r4 + u4 + d4 - 4.0f * c4);
        }
      }
      __syncthreads();
      float (*t)[TSTR] = src; src = dst; dst = t;
    }
  } else {
    // Exact clamp-aware scalar path; image-edge sides don't shrink.
    const int tx = tid & 31;
    const int ty = tid >> 5;
    for (int s = 0; s < NSTEPS; ++s) {
      const int lox = (bgx <= 0)        ? -bgx      : (s + 1);
      const int loy = (bgy <= 0)        ? -bgy      : (s + 1);
      const int hix = (bgx + TILE >= W) ? (W - bgx) : (TILE - (s + 1));
      const int hiy = (bgy + TILE >= H) ? (H - bgy) : (TILE - (s + 1));
      for (int ly = loy + ty; ly < hiy; ly += 8) {
        const int gy = bgy + ly;
        const int ym = ly - (gy > 0);
        const int yp = ly + (gy < H - 1);
        for (int lx = lox + tx; lx < hix; lx += 32) {
          const int gx = bgx + lx;
          const float c = src[ly][lx + CSH];
          const float l = src[ly][lx - (gx > 0) + CSH];
          const float r = src[ly][lx + (gx < W - 1) + CSH];
          const float u = src[ym][lx + CSH];
          const float d = src[yp][lx + CSH];
          dst[ly][lx + CSH] = c + ALPHA * (l + r + u + d - 4.0f * c);
        }
      }
      __syncthreads();
      float (*t)[TSTR] = src; src = dst; dst = t;
    }
  }
  // After 10 steps (even count), result lives in bufA == src.

  // ---- Phase 3: drain 64x64 output tile, b128 async LDS -> global ----
  // LDS col base HALO+CSH = 12 -> 48B (16B aligned); global gx = 64*bx + cx (16B aligned).
#if HAVE_AS128
  for (int i = tid; i < (BX * BX) / 4; i += 256) {
    int cx = (i & 15) * 4;
    int ly = i >> 4;
    int gx = (int)blockIdx.x * BX + cx;
    int gy = bgy + HALO + ly;
    __builtin_amdgcn_global_store_async_from_lds_b128(
        (g_v4i*)(oimg + (size_t)gy * W + gx),
        (l_v4i*)&src[ly + HALO][cx + HALO + CSH], 0, 0);
  }
  wait_async_zero();
#else
  for (int i = tid; i < BX * BX; i += 256) {
    int lx = i & (BX - 1);
    int ly = i >> 6;
    int gx = (int)blockIdx.x * BX + lx;
    int gy = bgy + HALO + ly;
    oimg[(size_t)gy * W + gx] = src[ly + HALO][lx + HALO + CSH];
  }
#endif
}

extern "C" void kernel_launch(void* const* d_in, const int* in_sizes, int n_in,
                              void* d_out, int out_size, void* d_ws, size_t ws_size,
                              hipStream_t stream) {
  const float* grid = (const float*)d_in[0];
  float* out = (float*)d_out;
  const int n_img = in_sizes[0] / (W * H);   // 8*4 = 32
  dim3 blocks(W / BX, H / BX, n_img);
  jacobi10_fused<<<blocks, 256, 0, stream>>>(grid, out);
}